// BinaryJumpingRNNCell_78735340471013
// MI455X (gfx1250) — compile-verified
//
#include <hip/hip_runtime.h>
#include <hip/hip_bf16.h>

typedef float v2f __attribute__((ext_vector_type(2)));
typedef float v8f __attribute__((ext_vector_type(8)));

#define B_   1024
#define T_   1024
#define D_   256
#define M_   10
#define LROW 258   // LDS row stride (floats): keeps float2 alignment, avoids bank conflicts

// ---- 16x16 f32 tile GEMM, A from LDS (row stride LROW), B = weight matrix in global ----
// Computes C[16x16] = A(16 x 256) x W(256 x 16-col-slice) with V_WMMA_F32_16X16X4_F32.
__device__ __forceinline__ v8f tile_mm_ldsA(const float* __restrict__ sA,
                                            const float* __restrict__ W,
                                            int n0, int g, int lr) {
  v8f acc = {0.f, 0.f, 0.f, 0.f, 0.f, 0.f, 0.f, 0.f};
  const float* arow = sA + lr * LROW + 2 * g;           // A[lr][k + 2g + j]
  const float* wcol = W + (size_t)(2 * g) * D_ + n0 + lr; // W[k + 2g + j][n0 + lr]
  #pragma unroll 4
  for (int k = 0; k < D_; k += 4) {
    v2f a = *(const v2f*)(arow + k);
    v2f b;
    b.x = wcol[(size_t)k * D_];
    b.y = wcol[(size_t)(k + 1) * D_];
    acc = __builtin_amdgcn_wmma_f32_16x16x4_f32(false, a, false, b,
                                                (short)0, acc, false, false);
  }
  return acc;
}

// ---- Same, but A rows come directly from global memory (gathered history rows) ----
// arow must already point at this lane's A row + 2*g element offset.
__device__ __forceinline__ v8f tile_mm_gA(const float* __restrict__ arow,
                                          const float* __restrict__ W,
                                          int n0, int g, int lr) {
  v8f acc = {0.f, 0.f, 0.f, 0.f, 0.f, 0.f, 0.f, 0.f};
  const float* wcol = W + (size_t)(2 * g) * D_ + n0 + lr;
  #pragma unroll 4
  for (int k = 0; k < D_; k += 4) {
    v2f a = *(const v2f*)(arow + k);
    v2f b;
    b.x = wcol[(size_t)k * D_];
    b.y = wcol[(size_t)(k + 1) * D_];
    acc = __builtin_amdgcn_wmma_f32_16x16x4_f32(false, a, false, b,
                                                (short)0, acc, false, false);
  }
  return acc;
}

__device__ __forceinline__ float sigmoidf_(float v) {
  return 1.f / (1.f + __expf(-v));
}

// ---- bulk copy: new_history = h_history (row `cur` overwritten later by cell kernel) ----
__global__ __launch_bounds__(256)
void hist_copy_kernel(const float4* __restrict__ src, float4* __restrict__ dst, size_t n4) {
  size_t i = (size_t)blockIdx.x * blockDim.x + threadIdx.x;
  size_t stride = (size_t)gridDim.x * blockDim.x;
  for (; i < n4; i += stride) dst[i] = src[i];
}

// ---- fused cell: q/K/V GEMMs + attention + h + y, 16 batch rows per block, 8 waves ----
__global__ __launch_bounds__(256, 1)
void cell_kernel(const float* __restrict__ x, const float* __restrict__ hist,
                 const float* __restrict__ Wi, const float* __restrict__ bi,
                 const float* __restrict__ Wk, const float* __restrict__ bk,
                 const float* __restrict__ Wv, const float* __restrict__ bv,
                 const float* __restrict__ Wb, const float* __restrict__ bb,
                 const int* __restrict__ jumps, const int* __restrict__ curp,
                 float* __restrict__ out_h, float* __restrict__ out_y,
                 float* __restrict__ out_hist) {
  __shared__ float sA[16 * LROW];   // x tile, later reused to hold h tile
  __shared__ float sq[16 * LROW];   // q tile
  __shared__ float ssc[16 * 16];    // scores [batch][m]
  __shared__ float sat[16 * 16];    // attn   [batch][m]

  const int tid  = threadIdx.x;
  const int wave = tid >> 5;
  const int lane = tid & 31;
  const int g    = lane >> 4;   // half-wave group
  const int lr   = lane & 15;   // row-in-A / col-in-B index
  const int b0   = blockIdx.x * 16;
  const int cur  = curp[0];

  // stage x tile (16 rows x 256 cols) into LDS
  {
    int row = tid >> 4;
    int c0  = (tid & 15) * 16;
    const float* src = x + (size_t)(b0 + row) * D_ + c0;
    float* dst = sA + row * LROW + c0;
    #pragma unroll
    for (int i = 0; i < 16; i += 4)
      *(float4*)(dst + i) = *(const float4*)(src + i);
  }
  ssc[tid] = 0.f;   // 256 entries, 256 threads
  __syncthreads();

  // ---- phase 1: q = x @ W_in + b_in  (kept in LDS) ----
  for (int nt = 2 * wave; nt < 2 * wave + 2; ++nt) {
    int n0 = nt * 16;
    v8f acc = tile_mm_ldsA(sA, Wi, n0, g, lr);
    int col = n0 + lr;
    float bias = bi[col];
    #pragma unroll
    for (int r = 0; r < 8; ++r)
      sq[(r + 8 * g) * LROW + col] = acc[r] + bias;
  }
  __syncthreads();

  // ---- phase 2: scores[b][m] = (q . (H_m @ W_k + b_k)) / 16 ----
  for (int m = 0; m < M_; ++m) {
    int t = cur - jumps[m]; if (t < 0) t = 0;
    const float* arow = hist + (size_t)(b0 + lr) * T_ * D_ + (size_t)t * D_ + 2 * g;
    for (int nt = 2 * wave; nt < 2 * wave + 2; ++nt) {
      int n0 = nt * 16;
      v8f acc = tile_mm_gA(arow, Wk, n0, g, lr);
      int col = n0 + lr;
      float bias = bk[col];
      #pragma unroll
      for (int r = 0; r < 8; ++r) {
        int bat = r + 8 * g;
        float contrib = (acc[r] + bias) * sq[bat * LROW + col] * 0.0625f; // 1/sqrt(256)
        atomicAdd(&ssc[bat * 16 + m], contrib);                           // ds_add_f32
      }
    }
  }
  __syncthreads();

  // ---- softmax over m (one thread per batch row) ----
  if (tid < 16) {
    float mx = -1e30f;
    for (int m = 0; m < M_; ++m) mx = fmaxf(mx, ssc[tid * 16 + m]);
    float e[M_], s = 0.f;
    for (int m = 0; m < M_; ++m) { e[m] = __expf(ssc[tid * 16 + m] - mx); s += e[m]; }
    float inv = 1.f / s;
    for (int m = 0; m < M_; ++m) sat[tid * 16 + m] = e[m] * inv;
  }
  __syncthreads();

  // ---- phase 3: rs = sum_m attn * (H_m @ W_v + b_v); h = sigmoid(q + rs) ----
  for (int nt = 2 * wave; nt < 2 * wave + 2; ++nt) {
    int n0 = nt * 16;
    int col = n0 + lr;
    float rs[8] = {0.f, 0.f, 0.f, 0.f, 0.f, 0.f, 0.f, 0.f};
    float bias = bv[col];
    for (int m = 0; m < M_; ++m) {
      int t = cur - jumps[m]; if (t < 0) t = 0;
      const float* arow = hist + (size_t)(b0 + lr) * T_ * D_ + (size_t)t * D_ + 2 * g;
      v8f acc = tile_mm_gA(arow, Wv, n0, g, lr);
      #pragma unroll
      for (int r = 0; r < 8; ++r) {
        int bat = r + 8 * g;
        rs[r] += sat[bat * 16 + m] * (acc[r] + bias);
      }
    }
    #pragma unroll
    for (int r = 0; r < 8; ++r) {
      int bat = r + 8 * g;
      float hv = sigmoidf_(sq[bat * LROW + col] + rs[r]);
      out_h[(size_t)(b0 + bat) * D_ + col] = hv;
      out_hist[((size_t)(b0 + bat) * T_ + cur) * D_ + col] = hv;
      sA[bat * LROW + col] = hv;   // reuse x buffer: h becomes A-matrix for y GEMM
    }
  }
  __syncthreads();

  // ---- phase 4: y = sigmoid(h @ W_bl + b_bl + h) ----
  for (int nt = 2 * wave; nt < 2 * wave + 2; ++nt) {
    int n0 = nt * 16;
    int col = n0 + lr;
    v8f acc = tile_mm_ldsA(sA, Wb, n0, g, lr);
    float bias = bb[col];
    #pragma unroll
    for (int r = 0; r < 8; ++r) {
      int bat = r + 8 * g;
      float pre = acc[r] + bias + sA[bat * LROW + col];
      out_y[(size_t)(b0 + bat) * D_ + col] = sigmoidf_(pre);
    }
  }
}

extern "C" void kernel_launch(void* const* d_in, const int* in_sizes, int n_in,
                              void* d_out, int out_size, void* d_ws, size_t ws_size,
                              hipStream_t stream) {
  (void)in_sizes; (void)n_in; (void)out_size; (void)d_ws; (void)ws_size;

  const float* x     = (const float*)d_in[0];
  const float* hist  = (const float*)d_in[1];
  const float* Wi    = (const float*)d_in[2];
  const float* bi    = (const float*)d_in[3];
  const float* Wk    = (const float*)d_in[4];
  const float* bk    = (const float*)d_in[5];
  const float* Wv    = (const float*)d_in[6];
  const float* bv    = (const float*)d_in[7];
  const float* Wb    = (const float*)d_in[8];
  const float* bb    = (const float*)d_in[9];
  const int*   jumps = (const int*)d_in[10];
  const int*   cur   = (const int*)d_in[11];

  float* out      = (float*)d_out;
  float* out_h    = out;                              // (B, D)
  float* out_y    = out + (size_t)B_ * D_;            // (B, D)
  float* out_hist = out + (size_t)2 * B_ * D_;        // (B, T, D)

  // 1) bandwidth-bound history copy (~2 GiB moved -> ~92 us floor at 23.3 TB/s)
  size_t n4 = (size_t)B_ * T_ * D_ / 4;
  hist_copy_kernel<<<4096, 256, 0, stream>>>((const float4*)hist, (float4*)out_hist, n4);

  // 2) fused WMMA cell (overwrites history row `cur` with h)
  cell_kernel<<<B_ / 16, 256, 0, stream>>>(x, hist, Wi, bi, Wk, bk, Wv, bv, Wb, bb,
                                           jumps, cur, out_h, out_y, out_hist);
}